// QMatmul_8246337208551
// MI455X (gfx1250) — compile-verified
//
#include <hip/hip_runtime.h>

// SpMM (CSR, fixed degree 16) x dense [N_COLS, 64] -> [N_ROWS, 64], fp32.
// Routed through V_WMMA_F32_16X16X4_F32 (exact fp32 matrix path on gfx1250).
//
// Tiling: one 128-thread block (4 waves) per 16-row tile. Wave w owns the
// 16-wide feature tile [16w, 16w+16). K-chain of 256 (16 rows x 16 edges) is
// processed as 64 chunks of K=4; chunk t carries edges 4t..4t+3 of row t/4,
// so the A fragment is block-diagonal (only row t/4 nonzero).
//
// VGPR layouts used (CDNA5 ISA 7.12.2):
//   A 16x4 f32 : lanes 0-15 -> K=0(V0),K=1(V1); lanes 16-31 -> K=2(V0),K=3(V1)
//   D 16x16 f32: VGPR j -> M = 8*(lane>>4)+j, N = lane&15
//   B 4x16 f32 : mirror of A (V0: K=0/K=2 by lane half, V1: K=1/K=3)
//
// All gather/store offsets are kept in unsigned 32-bit so the backend emits
// saddr-form global loads/stores (GVS: SGPR base + 32-bit VGPR offset)
// instead of per-lane 64-bit address arithmetic. (Gather index < 6.4M,
// output index < 6.4M -- both fit easily.)

typedef float v2f __attribute__((ext_vector_type(2)));
typedef float v8f __attribute__((ext_vector_type(8)));

#define F_DIM 64
#define ROWS_PER_BLOCK 16
#define EDGES_PER_ROW 16
#define EDGES_PER_BLOCK (ROWS_PER_BLOCK * EDGES_PER_ROW)  // 256

__global__ __launch_bounds__(128) void spmm_wmma_f32_kernel(
    const int* __restrict__ rowptr,   // [n_rows+1] (doc contract: int32)
    const int* __restrict__ col,      // [E]
    const float* __restrict__ value,  // [E]
    const float* __restrict__ other,  // [n_cols, 64]
    float* __restrict__ out,          // [n_rows, 64]
    int n_rows)
{
    __shared__ int   s_col[EDGES_PER_BLOCK];
    __shared__ float s_val[EDGES_PER_BLOCK];

    const int rb   = blockIdx.x;        // 16-row tile index
    const int tid  = threadIdx.x;       // 0..127
    const int lane = tid & 31;
    const int wv   = tid >> 5;          // F-tile 0..3

    // Stage this tile's 256 edges (col as int32, value) into LDS, shared by
    // all 4 waves. Fixed degree 16: row r's edges start at rowptr[r].
    for (int eb = tid; eb < EDGES_PER_BLOCK; eb += 128) {
        const int rloc = eb >> 4;
        const int k    = eb & 15;
        const int g    = rowptr[rb * ROWS_PER_BLOCK + rloc] + k;
        s_col[eb] = col[g];
        s_val[eb] = value[g];
    }
    __syncthreads();

    const unsigned half = lane >> 4;          // 0 -> K in {0,1}, 1 -> K in {2,3}
    const unsigned mrow = lane & 15;          // A row / D column within tile
    const unsigned ncol = wv * 16u + mrow;    // feature column for B and D

    v8f acc = {};                             // D/C accumulator, 16x16 f32

    #pragma unroll 8
    for (int t = 0; t < 64; ++t) {
        const unsigned r    = (unsigned)t >> 2;               // active row
        const unsigned kidx = ((unsigned)t << 2) + (half << 1);

        // A fragment: block-diagonal, only row r nonzero (v_cndmask -- no
        // EXEC divergence; WMMA requires EXEC all-ones).
        const float av0 = s_val[kidx];            // LDS broadcast per half
        const float av1 = s_val[kidx + 1];
        v2f a;
        a.x = (mrow == r) ? av0 : 0.0f;
        a.y = (mrow == r) ? av1 : 0.0f;

        // B fragment: gathered feature rows. `other` is 25.6 MB -> fully
        // L2-resident (192 MB); 16 lanes per half read 16 consecutive floats
        // = one 64B coalesced beat per gathered row. 32-bit offsets ->
        // saddr-mode global_load_b32.
        const unsigned c0 = (unsigned)s_col[kidx];
        const unsigned c1 = (unsigned)s_col[kidx + 1];
        v2f b;
        b.x = other[c0 * (unsigned)F_DIM + ncol];
        b.y = other[c1 * (unsigned)F_DIM + ncol];

        // D = A x B + C, full fp32 (8 args: neg_a, A, neg_b, B, c_mod, C,
        // reuse_a, reuse_b).
        acc = __builtin_amdgcn_wmma_f32_16x16x4_f32(
            false, a, false, b, (short)0, acc, false, false);
    }

    // Write D: VGPR j holds M = 8*half + j, N = mrow.
    const unsigned row0 = (unsigned)rb * ROWS_PER_BLOCK + half * 8u;
    #pragma unroll
    for (int j = 0; j < 8; ++j) {
        out[(row0 + (unsigned)j) * (unsigned)F_DIM + ncol] = acc[j];
    }
}

// Scalar tail for n_rows not divisible by 16 (unused for N_ROWS=100000, kept
// for generality; one thread per (row, feature) element).
__global__ void spmm_tail_kernel(
    const int* __restrict__ rowptr, const int* __restrict__ col,
    const float* __restrict__ value, const float* __restrict__ other,
    float* __restrict__ out, int row_start, int n_rows)
{
    const int idx = blockIdx.x * blockDim.x + threadIdx.x;
    const int row = row_start + idx / F_DIM;
    const int f   = idx % F_DIM;
    if (row >= n_rows) return;
    const int s = rowptr[row];
    const int e = rowptr[row + 1];
    float acc = 0.0f;
    for (int g = s; g < e; ++g)
        acc = fmaf(value[g], other[(size_t)col[g] * F_DIM + f], acc);
    out[(size_t)row * F_DIM + f] = acc;
}

extern "C" void kernel_launch(void* const* d_in, const int* in_sizes, int n_in,
                              void* d_out, int out_size, void* d_ws, size_t ws_size,
                              hipStream_t stream) {
    // setup_inputs() order: rowptr (int), col (int), value (f32), other (f32)
    const int*   rowptr = (const int*)d_in[0];
    const int*   col    = (const int*)d_in[1];
    const float* value  = (const float*)d_in[2];
    const float* other  = (const float*)d_in[3];
    float*       out    = (float*)d_out;

    const int n_rows  = in_sizes[0] - 1;           // 100000
    const int nblocks = n_rows / ROWS_PER_BLOCK;   // 6250 (exact)

    if (nblocks > 0) {
        spmm_wmma_f32_kernel<<<nblocks, 128, 0, stream>>>(
            rowptr, col, value, other, out, n_rows);
    }
    const int rem = n_rows - nblocks * ROWS_PER_BLOCK;
    if (rem > 0) {
        const int nt = rem * F_DIM;
        spmm_tail_kernel<<<(nt + 255) / 256, 256, 0, stream>>>(
            rowptr, col, value, other, out, nblocks * ROWS_PER_BLOCK, n_rows);
    }
}